// FrequencyMixingTABModule_66975720014061
// MI455X (gfx1250) — compile-verified
//
#include <hip/hip_runtime.h>
#include <math.h>

// ---------------- problem constants (from reference) ----------------
#define NN   128
#define CC   256
#define CZ   128
#define TT   8
#define NB   8
#define HH   19
#define WW   19
#define SS   (HH * WW)          // 361
#define C2   (2 * CZ)           // 256
#define MQ   (C2 * TT)          // 2048

typedef __attribute__((ext_vector_type(2))) float v2f;
typedef __attribute__((ext_vector_type(8))) float v8f;
typedef __attribute__((ext_vector_type(4))) int   v4i;

#if __has_builtin(__builtin_amdgcn_global_load_async_to_lds_b32) && \
    __has_builtin(__builtin_amdgcn_global_load_async_to_lds_b128) && \
    __has_builtin(__builtin_amdgcn_s_wait_asynccnt)
#define USE_ASYNC_LDS 1
#else
#define USE_ASYNC_LDS 0
#endif

#if USE_ASYNC_LDS
// Async global -> LDS copies (ASYNCcnt-tracked), typed per builtin signature:
// first arg is AS1 (global/__device__) pointer, second is AS3 (LDS) pointer.
__device__ __forceinline__ void async_g2l_b128(const void* g, void* l) {
    __builtin_amdgcn_global_load_async_to_lds_b128(
        (__attribute__((address_space(1))) v4i*)g,
        (__attribute__((address_space(3))) v4i*)l, 0, 0);
}
__device__ __forceinline__ void async_g2l_b32(const void* g, void* l) {
    __builtin_amdgcn_global_load_async_to_lds_b32(
        (__attribute__((address_space(1))) int*)g,
        (__attribute__((address_space(3))) int*)l, 0, 0);
}
#endif

// ------------------------------------------------------------------
// RoPE table: cos/sin of (f0*x + f1*y) per (cz, h, w)
// ------------------------------------------------------------------
__global__ void rope_kernel(const float* __restrict__ freqs,
                            float* __restrict__ cosb, float* __restrict__ sinb) {
    int idx = blockIdx.x * blockDim.x + threadIdx.x;
    if (idx >= CZ * SS) return;
    int p  = idx % SS;
    int cz = idx / SS;
    int h = p / WW, w = p % WW;
    float ang = freqs[cz * 2 + 0] * (float)w + freqs[cz * 2 + 1] * (float)h;
    cosb[idx] = __cosf(ang);
    sinb[idx] = __sinf(ang);
}

// ------------------------------------------------------------------
// Fused elementwise: optional scale+SiLU, optional rotation (sign rsgn).
// Layout (N, 2, CZ, S); one thread owns the (r,i) pair -> in-place safe.
// ------------------------------------------------------------------
__global__ void ew_kernel(const float* __restrict__ In, float* __restrict__ Out,
                          const float* __restrict__ cosb, const float* __restrict__ sinb,
                          float scale, int do_silu, int do_rot, float rsgn) {
    int idx = blockIdx.x * blockDim.x + threadIdx.x;
    if (idx >= NN * CZ * SS) return;
    int p  = idx % SS;
    int cz = (idx / SS) % CZ;
    int n  = idx / (SS * CZ);
    size_t ir = ((size_t)n * C2 + cz) * SS + p;          // real part channel
    size_t ii = ir + (size_t)CZ * SS;                    // imag part channel
    float r = In[ir] * scale;
    float i = In[ii] * scale;
    if (do_silu) {
        r = r * (1.0f / (1.0f + __expf(-r)));
        i = i * (1.0f / (1.0f + __expf(-i)));
    }
    if (do_rot) {
        float c = cosb[cz * SS + p];
        float s = rsgn * sinb[cz * SS + p];
        float nr = r * c - i * s;
        float ni = r * s + i * c;
        r = nr; i = ni;
    }
    Out[ir] = r;
    Out[ii] = i;
}

// ------------------------------------------------------------------
// Complex depthwise 3x3 conv with dilation (zero padding).
// ------------------------------------------------------------------
__global__ void cdw_kernel(const float* __restrict__ In, float* __restrict__ Out,
                           const float* __restrict__ Kre, const float* __restrict__ Kim,
                           int dil) {
    int idx = blockIdx.x * blockDim.x + threadIdx.x;
    if (idx >= NN * CZ * SS) return;
    int p  = idx % SS;
    int cz = (idx / SS) % CZ;
    int n  = idx / (SS * CZ);
    int h = p / WW, w = p % WW;
    size_t baseR = ((size_t)n * C2 + cz) * SS;
    size_t baseI = baseR + (size_t)CZ * SS;
    const float* kr = Kre + cz * 9;
    const float* ki = Kim + cz * 9;
    float or_ = 0.0f, oi_ = 0.0f;
#pragma unroll
    for (int dy = -1; dy <= 1; ++dy) {
#pragma unroll
        for (int dx = -1; dx <= 1; ++dx) {
            int hh = h + dy * dil;
            int ww = w + dx * dil;
            if (hh < 0 || hh >= HH || ww < 0 || ww >= WW) continue;
            float xr = In[baseR + hh * WW + ww];
            float xi = In[baseI + hh * WW + ww];
            float wr = kr[(dy + 1) * 3 + (dx + 1)];
            float wi = ki[(dy + 1) * 3 + (dx + 1)];
            or_ += xr * wr - xi * wi;
            oi_ += xr * wi + xi * wr;
        }
    }
    Out[baseR + p] = or_;
    Out[baseI + p] = oi_;
}

// ------------------------------------------------------------------
// Tiled WMMA fp32 GEMM:  Out[n,m,p] = sum_k W[m,k] * In[n,k,p]
// grid = (ceil(S/64), M/64, N); block = 128 threads = 4 waves.
// Each wave computes a 32x32 output tile (4 accumulators) via
// V_WMMA_F32_16X16X4_F32.  Double-buffered async global->LDS staging:
// chunk c+1 copies overlap chunk c's 64 WMMAs (ASYNCcnt in-order wait).
// Both LDS tiles are k-contiguous with padded stride 68 so every
// fragment is one conflict-free ds_load_b64.
// EPI: 0 none, 1 SiLU, 2 add skip (Aux[n,m,p]), 3 mask mul (Aux[n,p]).
// ------------------------------------------------------------------
template <int EPI>
__global__ __launch_bounds__(128)
void gemm_kernel(const float* __restrict__ Wm, const float* __restrict__ In,
                 float* __restrict__ Out, const float* __restrict__ Aux,
                 int M, int K) {
    constexpr int KC = 64;
    constexpr int PT = 64;
    constexpr int LDK = KC + 4;          // padded k-stride (68): conflict free
    __shared__ float lAs[2][64][LDK];    // [buf][m_local][k_local]
    __shared__ float lBs[2][PT][LDK];    // [buf][p_local][k_local]  (p-major!)

    const int n  = blockIdx.z;
    const int p0 = blockIdx.x * PT;
    const int m0 = blockIdx.y * 64;
    const int t    = threadIdx.x;
    const int wave = t >> 5;
    const int lane = t & 31;
    const int l16  = lane & 15;
    const int hi   = lane >> 4;          // 0 for lanes 0-15, 1 for lanes 16-31
    const int mw   = wave & 1;           // m sub-tile of the wave
    const int pw   = wave >> 1;          // p sub-tile of the wave

    const float* Wblk = Wm + (size_t)m0 * K;
    const float* Inp  = In + (size_t)n * K * SS;

    v8f acc[2][2] = {};

    const int arow0 = mw * 32 + l16;
    const int arow1 = arow0 + 16;
    const int bcol0 = pw * 32 + l16;
    const int bcol1 = bcol0 + 16;
    const int koff  = hi * 2;

    // Stage one K-chunk into LDS buffer `buf`.  Every thread issues exactly
    // 8 b128 (A) + 32 b32 (B) = 40 async ops; out-of-range B columns are
    // clamped (their garbage lands only in columns the epilogue discards).
    auto stage = [&](int kc, int buf) {
#if USE_ASYNC_LDS
        for (int i = t; i < 64 * (KC / 4); i += 128) {
            int mm = i >> 4;                 // / (KC/4)
            int k4 = (i & (KC / 4 - 1)) * 4;
            async_g2l_b128(Wblk + (size_t)mm * K + kc + k4, &lAs[buf][mm][k4]);
        }
        for (int i = t; i < KC * PT; i += 128) {
            int kk = i >> 6;                 // / PT
            int pp = i & (PT - 1);
            int p  = p0 + pp;
            if (p >= SS) p = SS - 1;
            async_g2l_b32(Inp + (size_t)(kc + kk) * SS + p, &lBs[buf][pp][kk]);
        }
#else
        for (int i = t; i < 64 * KC; i += 128) {
            int mm = i >> 6;
            int kk = i & (KC - 1);
            lAs[buf][mm][kk] = Wblk[(size_t)mm * K + kc + kk];
        }
        for (int i = t; i < KC * PT; i += 128) {
            int kk = i >> 6;
            int pp = i & (PT - 1);
            int p  = p0 + pp;
            if (p >= SS) p = SS - 1;
            lBs[buf][pp][kk] = Inp[(size_t)(kc + kk) * SS + p];
        }
#endif
    };

    const int nc = K / KC;
    stage(0, 0);
    for (int c = 0; c < nc; ++c) {
        const int buf = c & 1;
        if (c + 1 < nc) {
            stage((c + 1) * KC, 1 - buf);        // overlaps compute below
#if USE_ASYNC_LDS
            __builtin_amdgcn_s_wait_asynccnt(40); // chunk c complete (in-order)
#endif
        } else {
#if USE_ASYNC_LDS
            __builtin_amdgcn_s_wait_asynccnt(0);
#endif
        }
        __syncthreads();

        const float (*As)[LDK] = lAs[buf];
        const float (*Bs)[LDK] = lBs[buf];
#pragma unroll
        for (int k4 = 0; k4 < KC; k4 += 4) {
            const int ka = k4 + koff;
            v2f a0 = *(const v2f*)&As[arow0][ka];
            v2f a1 = *(const v2f*)&As[arow1][ka];
            v2f b0 = *(const v2f*)&Bs[bcol0][ka];
            v2f b1 = *(const v2f*)&Bs[bcol1][ka];
            acc[0][0] = __builtin_amdgcn_wmma_f32_16x16x4_f32(
                false, a0, false, b0, (short)0, acc[0][0], false, false);
            acc[0][1] = __builtin_amdgcn_wmma_f32_16x16x4_f32(
                false, a0, false, b1, (short)0, acc[0][1], false, false);
            acc[1][0] = __builtin_amdgcn_wmma_f32_16x16x4_f32(
                false, a1, false, b0, (short)0, acc[1][0], false, false);
            acc[1][1] = __builtin_amdgcn_wmma_f32_16x16x4_f32(
                false, a1, false, b1, (short)0, acc[1][1], false, false);
        }
        __syncthreads();   // safe to overwrite buf on iteration c+1's stage
    }

    // epilogue + store: VGPR r of tile (im,ip) holds
    //   M = m0 + mw*32 + im*16 + hi*8 + r,  P = p0 + pw*32 + ip*16 + l16
#pragma unroll
    for (int im = 0; im < 2; ++im) {
#pragma unroll
        for (int ip = 0; ip < 2; ++ip) {
            const int p = p0 + pw * 32 + ip * 16 + l16;
            if (p >= SS) continue;
            const int mbase = m0 + mw * 32 + im * 16 + hi * 8;
#pragma unroll
            for (int r = 0; r < 8; ++r) {
                float v = acc[im][ip][r];
                size_t oidx = ((size_t)n * M + (mbase + r)) * SS + p;
                if (EPI == 1) {
                    v = v * (1.0f / (1.0f + __expf(-v)));
                } else if (EPI == 2) {
                    v += Aux[oidx];
                } else if (EPI == 3) {
                    v *= Aux[(size_t)n * SS + p];
                }
                Out[oidx] = v;
            }
        }
    }
}

// ------------------------------------------------------------------
// In-place key rotation: layout (N, 2CZ, S), pair channels (2cz, 2cz+1).
// ------------------------------------------------------------------
__global__ void rot_keys_kernel(float* __restrict__ D,
                                const float* __restrict__ cosb,
                                const float* __restrict__ sinb, float sc) {
    int idx = blockIdx.x * blockDim.x + threadIdx.x;
    if (idx >= NN * CZ * SS) return;
    int p  = idx % SS;
    int cz = (idx / SS) % CZ;
    int n  = idx / (SS * CZ);
    size_t ir = ((size_t)n * C2 + 2 * cz) * SS + p;
    size_t ii = ir + SS;
    float r = D[ir], i = D[ii];
    float c = cosb[cz * SS + p];
    float s = sinb[cz * SS + p];
    D[ir] = (r * c - i * s) * sc;
    D[ii] = (r * s + i * c) * sc;
}

// ------------------------------------------------------------------
// In-place query rotation: layout (N, 2048, S), channel = (2cz+part)*T + t,
// so pair = (16cz + t, 16cz + 8 + t).
// ------------------------------------------------------------------
__global__ void rot_queries_kernel(float* __restrict__ D,
                                   const float* __restrict__ cosb,
                                   const float* __restrict__ sinb, float sc) {
    int idx = blockIdx.x * blockDim.x + threadIdx.x;
    if (idx >= NN * CZ * TT * SS) return;
    int p  = idx % SS;
    int tq = (idx / SS) % TT;
    int cz = (idx / (SS * TT)) % CZ;
    int n  = idx / (SS * TT * CZ);
    size_t ir = ((size_t)n * MQ + 16 * cz + tq) * SS + p;
    size_t ii = ir + (size_t)8 * SS;
    float r = D[ir], i = D[ii];
    float c = cosb[cz * SS + p];
    float s = sinb[cz * SS + p];
    D[ir] = (r * c - i * s) * sc;
    D[ii] = (r * s + i * c) * sc;
}

// ------------------------------------------------------------------
extern "C" void kernel_launch(void* const* d_in, const int* in_sizes, int n_in,
                              void* d_out, int out_size, void* d_ws, size_t ws_size,
                              hipStream_t stream) {
    const float* x       = (const float*)d_in[0];
    const float* mask    = (const float*)d_in[1];
    const float* w_in    = (const float*)d_in[2];
    const float* rope    = (const float*)d_in[3];
    const float* dw1_re  = (const float*)d_in[4];
    const float* dw1_im  = (const float*)d_in[5];
    const float* mix1    = (const float*)d_in[6];
    const float* dw2_re  = (const float*)d_in[7];
    const float* dw2_im  = (const float*)d_in[8];
    const float* mix2    = (const float*)d_in[9];
    const float* w_key   = (const float*)d_in[10];
    const float* w_query = (const float*)d_in[11];

    const size_t ZSZ = (size_t)NN * C2 * SS;   // 11,829,248 floats
    float* z    = (float*)d_ws;
    float* bufA = z + ZSZ;
    float* bufB = bufA + ZSZ;
    float* cosb = bufB + ZSZ;
    float* sinb = cosb + (size_t)CZ * SS;

    const int EWN = NN * CZ * SS;
    const dim3 ewg((EWN + 255) / 256);
    const dim3 ewb(256);
    const dim3 ggrid((SS + 63) / 64, C2 / 64, NN);
    const dim3 gblk(128);

    // RoPE tables
    rope_kernel<<<(CZ * SS + 255) / 256, 256, 0, stream>>>(rope, cosb, sinb);

    // z = (w_in @ x) * mask
    gemm_kernel<3><<<ggrid, gblk, 0, stream>>>(w_in, x, z, mask, C2, CC);

    for (int b = 0; b < NB; ++b) {
        const float inv = 1.0f / sqrtf((float)(b + 1));
        // silu(z/sqrt(b+1)) then rotate(+)
        ew_kernel<<<ewg, ewb, 0, stream>>>(z, bufA, cosb, sinb, inv, 1, 1, 1.0f);
        // complex dw conv, dil=2
        cdw_kernel<<<ewg, ewb, 0, stream>>>(bufA, bufB, dw1_re + (size_t)b * CZ * 9,
                                            dw1_im + (size_t)b * CZ * 9, 2);
        // rotate(-), in place
        ew_kernel<<<ewg, ewb, 0, stream>>>(bufB, bufB, cosb, sinb, 1.0f, 0, 1, -1.0f);
        // mix1 GEMM + SiLU epilogue
        gemm_kernel<1><<<ggrid, gblk, 0, stream>>>(mix1 + (size_t)b * C2 * C2, bufB,
                                                   bufA, nullptr, C2, C2);
        // rotate(+), in place
        ew_kernel<<<ewg, ewb, 0, stream>>>(bufA, bufA, cosb, sinb, 1.0f, 0, 1, 1.0f);
        // complex dw conv, dil=1
        cdw_kernel<<<ewg, ewb, 0, stream>>>(bufA, bufB, dw2_re + (size_t)b * CZ * 9,
                                            dw2_im + (size_t)b * CZ * 9, 1);
        // rotate(-), in place
        ew_kernel<<<ewg, ewb, 0, stream>>>(bufB, bufB, cosb, sinb, 1.0f, 0, 1, -1.0f);
        // mix2 GEMM + residual add (reads skip z, writes z in place per element)
        gemm_kernel<2><<<ggrid, gblk, 0, stream>>>(mix2 + (size_t)b * C2 * C2, bufB,
                                                   z, z, C2, C2);
    }

    // final activation: silu(z/3), no rotation
    ew_kernel<<<ewg, ewb, 0, stream>>>(z, bufA, cosb, sinb, 1.0f / 3.0f, 1, 0, 1.0f);

    float* keys    = (float*)d_out;                       // (N, 2CZ, 1, S)
    float* queries = keys + (size_t)NN * C2 * SS;         // (N, 2CZ, T, S)

    // key / query GEMMs straight into d_out, then rotate in place
    gemm_kernel<0><<<ggrid, gblk, 0, stream>>>(w_key, bufA, keys, nullptr, C2, C2);
    const dim3 qgrid((SS + 63) / 64, MQ / 64, NN);
    gemm_kernel<0><<<qgrid, gblk, 0, stream>>>(w_query, bufA, queries, nullptr, MQ, C2);

    rot_keys_kernel<<<ewg, ewb, 0, stream>>>(keys, cosb, sinb,
                                             1.0f / sqrtf((float)CZ));
    const int QN = NN * CZ * TT * SS;
    rot_queries_kernel<<<(QN + 255) / 256, 256, 0, stream>>>(queries, cosb, sinb,
                                                             1.0f / sqrtf((float)TT));
}